// Local_contrast_normalization_54271206752400
// MI455X (gfx1250) — compile-verified
//
#include <hip/hip_runtime.h>
#include <hip/hip_bf16.h>
#include <math.h>

// Local contrast normalization, fused single kernel, CDNA5 (gfx1250).
// Both 9x9 Gaussian convolutions are separable and implemented as banded-
// matrix multiplies on V_WMMA_F32_16X16X4_F32 (native fp32 matrix op).
// Wave-uniform tile loops are forced onto the scalar path (readfirstlane)
// so EXEC is all-ones around every WMMA, as the ISA requires.

typedef __attribute__((ext_vector_type(2))) float v2f;
typedef __attribute__((ext_vector_type(8))) float v8f;

#define IMG_H 512
#define IMG_W 512
#define TS    64          // output tile per workgroup
#define FS    80          // field = TS + 2*8 halo
#define XW    88          // xsh row width  = FS + 2*4 zero guard cols
#define T1H   88          // t1  row count  = FS + 2*4 zero guard rows
#define THRSH 0.5f

__global__ __launch_bounds__(256) void lcn_wmma_kernel(const float* __restrict__ x,
                                                       float* __restrict__ out) {
  // LDS: 28160 + 28160 + 25600 = 81920 B  -> 4 workgroups / WGP (320 KB)
  __shared__ float xsh[FS][XW];   // input tile (later: squared demeaned tile)
  __shared__ float t1b[T1H][FS];  // h-pass result (later: second h-pass result)
  __shared__ float dbf[FS][FS];   // demeaned values

  const int tid  = threadIdx.x;
  // wave id is wave-uniform: pin it to an SGPR so all tile loops are scalar
  // control flow and EXEC stays all-ones around the WMMAs.
  const int wv   = __builtin_amdgcn_readfirstlane(tid >> 5);
  const int lane = tid & 31;
  const int lm   = lane & 15;
  const int koff = (lane < 16) ? 0 : 2;  // K-pair held by this half-wave (A & B f32 layouts)
  const int mb   = (lane < 16) ? 0 : 8;  // C/D row offset for this half-wave
  const int gr0  = blockIdx.y * TS;
  const int gc0  = blockIdx.x * TS;
  const float* xb = x   + (size_t)blockIdx.z * (IMG_H * IMG_W);
  float*       ob = out + (size_t)blockIdx.z * (IMG_H * IMG_W);

  // ---- normalized 1-D Gaussian taps: g[i] ~ exp(-(i-4.5)^2 / (2*(9/6))) ----
  float gs = 0.f;
#pragma unroll
  for (int i = 0; i < 9; ++i) {
    float d = (float)i - 4.5f;
    gs += expf(-d * d * (1.0f / 3.0f));
  }
  const float ginv = 1.0f / gs;

  // Banded-filter WMMA operand.  W[m][u] = g[u-m], u in 0..23 split into six
  // 4-wide K chunks.  Per-lane storage is IDENTICAL for the B operand of the
  // horizontal pass (B_j[k][n] = g[4j+k-n]) and the A operand of the vertical
  // pass (A_j[m][k] = g[4j+k-m]): component x holds K=koff, y holds K=koff+1.
  v2f wmat[6];
#pragma unroll
  for (int j = 0; j < 6; ++j) {
    int   t0 = 4 * j + koff - lm;
    float d0 = (float)t0 - 4.5f;
    float d1 = (float)(t0 + 1) - 4.5f;
    wmat[j].x = (t0 >= 0 && t0 <= 8)         ? expf(-d0 * d0 * (1.0f / 3.0f)) * ginv : 0.0f;
    wmat[j].y = (t0 + 1 >= 0 && t0 + 1 <= 8) ? expf(-d1 * d1 * (1.0f / 3.0f)) * ginv : 0.0f;
  }

  // Horizontal conv of a 16x16 tile: D = sum_j A_j x B_j,  A_j = image chunk.
  auto hpass = [&](int r0, int c0) {
    v8f acc = {0.f, 0.f, 0.f, 0.f, 0.f, 0.f, 0.f, 0.f};
#pragma unroll
    for (int j = 0; j < 6; ++j) {
      v2f a = *(const v2f*)&xsh[r0 + lm][c0 + 4 * j + koff];  // 8B aligned
      acc = __builtin_amdgcn_wmma_f32_16x16x4_f32(
          false, a, false, wmat[j], (short)0, acc, false, false);
    }
#pragma unroll
    for (int i = 0; i < 8; ++i)
      t1b[r0 + mb + i + 4][c0 + lm] = acc[i];
  };

  // Vertical conv of a 16x16 tile: D = sum_j A_j x B_j,  B_j = column strip.
  auto vpass = [&](int r0, int c0) -> v8f {
    v8f acc = {0.f, 0.f, 0.f, 0.f, 0.f, 0.f, 0.f, 0.f};
#pragma unroll
    for (int j = 0; j < 6; ++j) {
      v2f b;
      b.x = t1b[r0 + 4 * j + koff][c0 + lm];
      b.y = t1b[r0 + 4 * j + koff + 1][c0 + lm];
      acc = __builtin_amdgcn_wmma_f32_16x16x4_f32(
          false, wmat[j], false, b, (short)0, acc, false, false);
    }
    return acc;
  };

  // ---- Phase 0: load 80x80 halo tile (SAME zero pad) + zero guards --------
  {
    // t1 guard rows 0..3 and 84..87 are two contiguous 320-float spans.
    float* g0 = &t1b[0][0];
    float* g1 = &t1b[FS + 4][0];
    for (int i = tid; i < 4 * FS; i += 256) { g0[i] = 0.0f; g1[i] = 0.0f; }
    // xsh guard cols 0..3 and 84..87 (power-of-two indexing, no div).
    for (int i = tid; i < FS * 8; i += 256) {
      int r = i >> 3, c = i & 7;
      xsh[r][(c < 4) ? c : (c + FS)] = 0.0f;
    }
    // main tile: row/col nest, no div/mod, coalesced along columns.
    const int tx = tid & 31, ty = tid >> 5;
    for (int r = ty; r < FS; r += 8) {
      for (int c = tx; c < FS; c += 32) {
        int rr = gr0 + r - 8, cc = gc0 + c - 8;
        float v = 0.0f;
        if ((unsigned)rr < IMG_H && (unsigned)cc < IMG_W) v = xb[rr * IMG_W + cc];
        xsh[r][c + 4] = v;
      }
    }
  }
  __syncthreads();

  // ---- Phase 1: horizontal conv of x over full 80x80 field (25 tiles) -----
  for (int t = wv; t < 25; t += 8) {
    int r0 = (t / 5) * 16, c0 = (t % 5) * 16;  // scalar (SGPR) arithmetic
    hpass(r0, c0);
  }
  __syncthreads();

  // ---- Phase 2: vertical conv -> local mean; demean, square (25 tiles) ----
  for (int t = wv; t < 25; t += 8) {
    int r0 = (t / 5) * 16, c0 = (t % 5) * 16;
    v8f m = vpass(r0, c0);
    int fc = c0 + lm;
#pragma unroll
    for (int i = 0; i < 8; ++i) {
      int   fr = r0 + mb + i;
      float xv = xsh[fr][fc + 4];
      float dv = xv - m[i];
      dbf[fr][fc] = dv;
      int ir = gr0 + fr - 8, ic = gc0 + fc - 8;
      bool inimg = ((unsigned)ir < IMG_H) && ((unsigned)ic < IMG_W);
      xsh[fr][fc + 4] = inimg ? dv * dv : 0.0f;  // reference zero-pads conv2 input
    }
  }
  __syncthreads();

  // ---- Phase 3: horizontal conv of d^2 (rows 0..79, cols 8..71: 20 tiles) -
  for (int t = wv; t < 20; t += 8) {
    int r0 = (t >> 2) * 16, c0 = 8 + (t & 3) * 16;
    hpass(r0, c0);
  }
  __syncthreads();

  // ---- Phase 4: vertical conv -> variance; normalize + store (16 tiles) ---
  for (int t = wv; t < 16; t += 8) {
    int r0 = 8 + (t >> 2) * 16, c0 = 8 + (t & 3) * 16;
    v8f vr = vpass(r0, c0);
    int fc = c0 + lm;
#pragma unroll
    for (int i = 0; i < 8; ++i) {
      int   fr  = r0 + mb + i;
      float dv  = dbf[fr][fc];
      float nrm = sqrtf(fmaxf(vr[i], 0.0f));
      float res = (nrm > THRSH) ? dv / nrm : dv;
      int ir = gr0 + fr - 8, ic = gc0 + fc - 8;   // always in-bounds (512 % 64 == 0)
      ob[ir * IMG_W + ic] = res;
    }
  }
}

extern "C" void kernel_launch(void* const* d_in, const int* in_sizes, int n_in,
                              void* d_out, int out_size, void* d_ws, size_t ws_size,
                              hipStream_t stream) {
  (void)n_in; (void)out_size; (void)d_ws; (void)ws_size;
  const float* x  = (const float*)d_in[0];
  float*       o  = (float*)d_out;
  int batch = in_sizes[0] / (IMG_H * IMG_W);   // 64 for the reference shapes
  dim3 grid(IMG_W / TS, IMG_H / TS, batch);    // 8 x 8 x 64
  lcn_wmma_kernel<<<grid, 256, 0, stream>>>(x, o);
}